// ReXGNN_19507741458590
// MI455X (gfx1250) — compile-verified
//
#include <hip/hip_runtime.h>
#include <math.h>

// Problem constants (from reference)
#define B_   16
#define T_   12
#define N_   5000
#define E_   80000
#define H_   64
#define OUT_ 12
#define BT_  (B_ * T_)
#define BTN_ (BT_ * N_)
#define SEQ_ (B_ * N_)      // 80000 GRU sequences

// LDS row stride (dwords) for 64-wide rows: (4n + 4kt + delta) mod 64 covers
// all 64 banks exactly once for a wave of b64 loads -> conflict-free.
#define STRD 68

typedef float v2f __attribute__((ext_vector_type(2)));
typedef float v8f __attribute__((ext_vector_type(8)));

// ---------------------------------------------------------------------------
// K0: rank-1 precompute.
//   u[g] = W_ih[g,:]·gcn_w,  v[g] = W_ih[g,:]·gcn_b   (collapses gi matmul)
//   scal = {p1,q1,p2,q2,tb}: attention-score affine coefficients.
// ---------------------------------------------------------------------------
__global__ __launch_bounds__(256) void k_precompute(
    const float* __restrict__ gcn_w, const float* __restrict__ gcn_b,
    const float* __restrict__ sa_w,  const float* __restrict__ sa_b,
    const float* __restrict__ ta_w,  const float* __restrict__ ta_b,
    const float* __restrict__ w_ih,  const float* __restrict__ b_ih,
    const float* __restrict__ b_hh,
    float* __restrict__ scal, float4* __restrict__ guv)
{
    int g = blockIdx.x * blockDim.x + threadIdx.x;
    if (g < 3 * H_) {
        float u = 0.f, v = 0.f;
        for (int k = 0; k < H_; ++k) {
            float w = w_ih[g * H_ + k];
            u += w * gcn_w[k];
            v += w * gcn_b[k];
        }
        guv[g] = make_float4(u, v, b_ih[g], b_hh[g]);
    }
    if (g == 0) {
        float p1 = 0.f, q1 = 0.f, p2 = 0.f, q2 = 0.f;
        for (int k = 0; k < H_; ++k) {
            p1 += gcn_w[k] * sa_w[k];
            q1 += gcn_b[k] * sa_w[k];
            p2 += gcn_w[k] * ta_w[k];
            q2 += gcn_b[k] * ta_w[k];
        }
        scal[0] = p1; scal[1] = q1 + sa_b[0];
        scal[2] = p2; scal[3] = q2; scal[4] = ta_b[0];
    }
}

// ---------------------------------------------------------------------------
// GCN degree / normalization (deg includes self-loop -> init 1.0)
// ---------------------------------------------------------------------------
__global__ __launch_bounds__(256) void k_deg_init(float* __restrict__ degf) {
    int n = blockIdx.x * blockDim.x + threadIdx.x;
    if (n < N_) degf[n] = 1.0f;
}

__global__ __launch_bounds__(256) void k_deg_acc(const int* __restrict__ ei,
                                                 float* __restrict__ degf) {
    int e = blockIdx.x * blockDim.x + threadIdx.x;
    if (e < E_) atomicAdd(&degf[ei[E_ + e]], 1.0f);
}

__global__ __launch_bounds__(256) void k_dinv(float* __restrict__ degf) {
    int n = blockIdx.x * blockDim.x + threadIdx.x;
    if (n < N_) degf[n] = rsqrtf(degf[n]);   // deg >= 1 always (self loop)
}

// agg init = self-loop contribution: x * dinv^2
__global__ __launch_bounds__(256) void k_agg_init(const float* __restrict__ x,
                                                  const float* __restrict__ dinv,
                                                  float* __restrict__ agg) {
    int i = blockIdx.x * blockDim.x + threadIdx.x;
    if (i < BTN_) {
        int n = i % N_;
        float di = dinv[n];
        agg[i] = x[i] * di * di;
    }
}

// edge scatter: one thread per (bt, e)
__global__ __launch_bounds__(256) void k_agg_edges(const float* __restrict__ x,
                                                   const int* __restrict__ ei,
                                                   const float* __restrict__ dinv,
                                                   float* __restrict__ agg) {
    unsigned int idx = blockIdx.x * blockDim.x + threadIdx.x;
    if (idx >= (unsigned int)(BT_ * E_)) return;
    int bt = idx / E_;
    int e  = idx % E_;
    int s = ei[e], d = ei[E_ + e];
    float v = x[bt * N_ + s] * dinv[s] * dinv[d];
    atomicAdd(&agg[bt * N_ + d], v);
}

// ---------------------------------------------------------------------------
// Spatial softmax over N per (b,t). score = agg*p1 + q1 (affine in agg).
// Emits c1 = alpha*agg, d1 = alpha, and temporal score ts = c1*p2 + d1*q2 + tb.
// ---------------------------------------------------------------------------
__global__ __launch_bounds__(256) void k_spatial(const float* __restrict__ agg,
                                                 const float* __restrict__ scal,
                                                 float* __restrict__ c1,
                                                 float* __restrict__ d1,
                                                 float* __restrict__ ts)
{
    __shared__ float red[256];
    const int bt = blockIdx.x;
    const float p1 = scal[0], q1 = scal[1], p2 = scal[2], q2 = scal[3], tb = scal[4];
    const float* a = agg + bt * N_;

    float m = -3.4e38f;
    for (int n = threadIdx.x; n < N_; n += 256) m = fmaxf(m, a[n] * p1 + q1);
    red[threadIdx.x] = m; __syncthreads();
    for (int s = 128; s > 0; s >>= 1) {
        if (threadIdx.x < s) red[threadIdx.x] = fmaxf(red[threadIdx.x], red[threadIdx.x + s]);
        __syncthreads();
    }
    const float smax = red[0]; __syncthreads();

    float sum = 0.f;
    for (int n = threadIdx.x; n < N_; n += 256) sum += expf(a[n] * p1 + q1 - smax);
    red[threadIdx.x] = sum; __syncthreads();
    for (int s = 128; s > 0; s >>= 1) {
        if (threadIdx.x < s) red[threadIdx.x] += red[threadIdx.x + s];
        __syncthreads();
    }
    const float inv = 1.f / red[0];

    for (int n = threadIdx.x; n < N_; n += 256) {
        float av = a[n];
        float alpha = expf(av * p1 + q1 - smax) * inv;
        float c = alpha * av, d = alpha;
        int i = bt * N_ + n;
        c1[i] = c; d1[i] = d;
        ts[i] = c * p2 + d * q2 + tb;
    }
}

// ---------------------------------------------------------------------------
// Temporal softmax over T=12 per (b,n); emit interleaved (c,d) per seq*step:
//   cd[s*24 + 2t] = beta*c1, cd[s*24 + 2t + 1] = beta*d1,  s = b*N + n.
// ---------------------------------------------------------------------------
__global__ __launch_bounds__(256) void k_temporal(const float* __restrict__ ts,
                                                  const float* __restrict__ c1,
                                                  const float* __restrict__ d1,
                                                  float* __restrict__ cd)
{
    int idx = blockIdx.x * blockDim.x + threadIdx.x;   // s = b*N + n
    if (idx >= SEQ_) return;
    int b = idx / N_, n = idx - b * N_;

    float tv[T_];
    float mx = -3.4e38f;
    for (int t = 0; t < T_; ++t) {
        float v = ts[(b * T_ + t) * N_ + n];
        tv[t] = v;
        mx = fmaxf(mx, v);
    }
    float sum = 0.f;
    for (int t = 0; t < T_; ++t) { tv[t] = expf(tv[t] - mx); sum += tv[t]; }
    float inv = 1.f / sum;
    for (int t = 0; t < T_; ++t) {
        float beta = tv[t] * inv;
        int i = (b * T_ + t) * N_ + n;
        cd[idx * 24 + 2 * t]     = beta * c1[i];
        cd[idx * 24 + 2 * t + 1] = beta * d1[i];
    }
}

// ---------------------------------------------------------------------------
// GRU over T=12 with V_WMMA_F32_16X16X4_F32 for gh = h @ W_hh^T, fused output
// projection. One wave32 = 16 sequences; 8 waves/block share W_hh in LDS.
//   A (16x4 f32): lane = M row (lane%16); VGPR0/1 = K {0,1} (lanes<16) or
//                 {2,3} (lanes>=16)  -> b64 from padded h tile.
//   B (4x16 f32): lane = N col (lane%16); VGPR0/1 = K {0,1}/{2,3} -> b64 from
//                 W_hh stored row-major [g][k] with stride 68 (bank-perfect).
//   C/D (16x16):  lane%16 = gate col, VGPR v = seq v (+8 for lanes>=16).
// ---------------------------------------------------------------------------
__global__ __launch_bounds__(256) void k_gru_wmma(
    const float*  __restrict__ w_hh,   // [192,64]
    const float*  __restrict__ cd,     // [SEQ_,12,2]
    const float4* __restrict__ guv,    // [192] (u, v, b_ih, b_hh)
    const float*  __restrict__ proj_w, // [64,12]
    const float*  __restrict__ proj_b, // [12]
    float*        __restrict__ out)    // [B,OUT,N]
{
    extern __shared__ float smem[];
    float* whh_lds = smem;                       // 192*68
    float* h_all   = smem + 192 * STRD;          // 8*16*68
    float* cd_all  = h_all + 8 * 16 * STRD;      // 8*16*24

    const int tid  = threadIdx.x;
    const int wave = tid >> 5;       // wave32
    const int lane = tid & 31;
    const int ln   = lane & 15;
    const int hi   = lane >> 4;      // half-wave select (K or M offset)
    const int s0   = blockIdx.x * 128 + wave * 16;

    // Cooperative load of W_hh into padded LDS (read-only thereafter)
    for (int i = tid; i < 192 * 64; i += 256) {
        int g = i >> 6, k = i & 63;
        whh_lds[g * STRD + k] = w_hh[i];
    }
    float* hrow = h_all + wave * 16 * STRD;
    float* cd_w = cd_all + wave * 16 * 24;
    for (int i = lane; i < 16 * 24; i += 32) cd_w[i] = cd[s0 * 24 + i];
    for (int i = lane; i < 16 * STRD; i += 32) hrow[i] = 0.f;   // h0 = 0

    // Time-invariant per-gate constants, gate g = 16*nt + ln
    float u_[12], vv_[12], bi_[12], bh_[12];
    #pragma unroll
    for (int nt = 0; nt < 12; ++nt) {
        const float4 q = guv[16 * nt + ln];
        u_[nt] = q.x; vv_[nt] = q.y; bi_[nt] = q.z; bh_[nt] = q.w;
    }
    __syncthreads();

    v8f hreg[4] = {};   // h in C/D layout: hreg[jt][v] = h[seq v+8*hi][16*jt+ln]

    #pragma unroll 1
    for (int t = 0; t < T_; ++t) {
        v8f acc[12] = {};   // gh accumulators, 12 gate tiles of 16
        #pragma unroll
        for (int kt = 0; kt < 16; ++kt) {
            v2f a = *(const v2f*)(hrow + ln * STRD + 4 * kt + 2 * hi);
            #pragma unroll
            for (int nt = 0; nt < 12; ++nt) {
                v2f bm = *(const v2f*)(whh_lds + (16 * nt + ln) * STRD + 4 * kt + 2 * hi);
                acc[nt] = __builtin_amdgcn_wmma_f32_16x16x4_f32(
                    false, a, false, bm, (short)0, acc[nt], false, false);
            }
        }
        // Elementwise GRU update; gi reconstructed from rank-1 coefficients
        #pragma unroll
        for (int v = 0; v < 8; ++v) {
            const int m = v + 8 * hi;                       // sequence in tile
            v2f cdv = *(const v2f*)(cd_w + m * 24 + 2 * t);
            const float c = cdv.x, d = cdv.y;
            #pragma unroll
            for (int jt = 0; jt < 4; ++jt) {                // hidden unit 16*jt+ln
                float gi_r = c * u_[jt]     + d * vv_[jt]     + bi_[jt];
                float gi_z = c * u_[4 + jt] + d * vv_[4 + jt] + bi_[4 + jt];
                float gi_n = c * u_[8 + jt] + d * vv_[8 + jt] + bi_[8 + jt];
                float gh_r = acc[jt][v]     + bh_[jt];
                float gh_z = acc[4 + jt][v] + bh_[4 + jt];
                float gh_n = acc[8 + jt][v] + bh_[8 + jt];
                float r  = 1.f / (1.f + expf(-(gi_r + gh_r)));
                float z  = 1.f / (1.f + expf(-(gi_z + gh_z)));
                float ng = tanhf(gi_n + r * gh_n);
                float hn = (1.f - z) * ng + z * hreg[jt][v];
                hreg[jt][v] = hn;
                hrow[m * STRD + 16 * jt + ln] = hn;         // A-operand for t+1
            }
        }
    }

    // Fused output projection: out[b][o][n] = h_last · proj_w[:,o] + proj_b[o]
    for (int idx = lane; idx < 16 * OUT_; idx += 32) {
        int m = idx / OUT_, o = idx - m * OUT_;
        float sum = proj_b[o];
        #pragma unroll 8
        for (int k = 0; k < H_; ++k)
            sum += hrow[m * STRD + k] * proj_w[k * OUT_ + o];
        int s = s0 + m;
        int b = s / N_, n = s - b * N_;
        out[(b * OUT_ + o) * N_ + n] = sum;
    }
}

// ---------------------------------------------------------------------------
// Host-side launcher
// ---------------------------------------------------------------------------
extern "C" void kernel_launch(void* const* d_in, const int* in_sizes, int n_in,
                              void* d_out, int out_size, void* d_ws, size_t ws_size,
                              hipStream_t stream) {
    const float* x      = (const float*)d_in[0];
    const int*   ei     = (const int*)  d_in[1];
    const float* gcn_w  = (const float*)d_in[2];
    const float* gcn_b  = (const float*)d_in[3];
    const float* sa_w   = (const float*)d_in[4];
    const float* sa_b   = (const float*)d_in[5];
    const float* ta_w   = (const float*)d_in[6];
    const float* ta_b   = (const float*)d_in[7];
    const float* w_ih   = (const float*)d_in[8];
    const float* w_hh   = (const float*)d_in[9];
    const float* b_ih   = (const float*)d_in[10];
    const float* b_hh   = (const float*)d_in[11];
    const float* proj_w = (const float*)d_in[12];
    const float* proj_b = (const float*)d_in[13];
    float* out = (float*)d_out;
    float* ws  = (float*)d_ws;

    // Workspace layout (floats); total ~23 MB
    float*  scal = ws;                    // 8
    float4* guv  = (float4*)(ws + 8);     // 192 float4
    float*  dinv = ws + 1024;             // N (deg -> rsqrt in place)
    float*  agg  = ws + 8192;             // BT*N
    float*  c1   = agg + BTN_;            // BT*N
    float*  d1   = c1 + BTN_;             // BT*N
    float*  ts   = d1 + BTN_;             // BT*N
    float*  cd   = ts + BTN_;             // SEQ_*24

    k_precompute<<<1, 256, 0, stream>>>(gcn_w, gcn_b, sa_w, sa_b, ta_w, ta_b,
                                        w_ih, b_ih, b_hh, scal, guv);
    k_deg_init<<<(N_ + 255) / 256, 256, 0, stream>>>(dinv);
    k_deg_acc<<<(E_ + 255) / 256, 256, 0, stream>>>(ei, dinv);
    k_dinv<<<(N_ + 255) / 256, 256, 0, stream>>>(dinv);
    k_agg_init<<<(BTN_ + 255) / 256, 256, 0, stream>>>(x, dinv, agg);
    k_agg_edges<<<(BT_ * E_ + 255) / 256, 256, 0, stream>>>(x, ei, dinv, agg);
    k_spatial<<<BT_, 256, 0, stream>>>(agg, scal, c1, d1, ts);
    k_temporal<<<(SEQ_ + 255) / 256, 256, 0, stream>>>(ts, c1, d1, cd);

    const size_t lds_bytes = (size_t)(192 * STRD + 8 * 16 * STRD + 8 * 16 * 24) * 4;
    k_gru_wmma<<<SEQ_ / 128, 256, lds_bytes, stream>>>(w_hh, cd, guv,
                                                       proj_w, proj_b, out);
}